// YoloLoss_188978561261
// MI455X (gfx1250) — compile-verified
//
#include <hip/hip_runtime.h>
#include <cstdint>
#include <cstddef>

#define BLK 256

// ---------------------------------------------------------------------------
// CDNA5 async global->LDS copy (one dword per lane), tracked by ASYNCcnt.
// ---------------------------------------------------------------------------
__device__ __forceinline__ void async_ld_b32(const float* g, float* l) {
#if defined(__has_builtin) && __has_builtin(__builtin_amdgcn_global_load_async_to_lds_b32)
    __builtin_amdgcn_global_load_async_to_lds_b32(
        (__attribute__((address_space(1))) int*)(g),
        (__attribute__((address_space(3))) int*)(l),
        /*offset=*/0, /*cpol=*/0);
#else
    unsigned lds_off = (unsigned)(size_t)(__attribute__((address_space(3))) float*)l;
    asm volatile("global_load_async_to_lds_b32 %0, %1, off"
                 :: "v"(lds_off), "v"(g) : "memory");
#endif
}

template <int N>
__device__ __forceinline__ void wait_asynccnt() {
#if defined(__has_builtin) && __has_builtin(__builtin_amdgcn_s_wait_asynccnt)
    __builtin_amdgcn_s_wait_asynccnt(N);
#else
    asm volatile("s_wait_asynccnt %0" :: "i"(N) : "memory");
#endif
}

__device__ __forceinline__ float bce_logits(float x, float t) {
    return fmaxf(x, 0.f) - x * t + log1pf(expf(-fabsf(x)));
}

// ---------------------------------------------------------------------------
// Workspace layout (floats):
//   [0,307200)        tobj level 0
//   [307200,384000)   tobj level 1
//   [384000,403200)   tobj level 2
//   [403200,403209)   per-level {lbox_sum, lcls_sum, nvalid}
//   [403216,403472)   lobj partials level0 (256)
//   [403472,403600)   lobj partials level1 (128)
//   [403600,403664)   lobj partials level2 (64)
// ---------------------------------------------------------------------------
#define TOBJ0 0
#define TOBJ1 307200
#define TOBJ2 384000
#define ACC   403200
#define P0OFF 403216
#define P1OFF 403472
#define P2OFF 403600

__global__ void zero_kernel(float* __restrict__ ws, int n) {
    int i = blockIdx.x * blockDim.x + threadIdx.x;
    int stride = gridDim.x * blockDim.x;
    for (; i < n; i += stride) ws[i] = 0.f;
}

// One block per pyramid level; 900 candidate rows each.
__global__ __launch_bounds__(1024)
void match_kernel(const float* __restrict__ p0, const float* __restrict__ p1,
                  const float* __restrict__ p2, const float* __restrict__ targets,
                  const float* __restrict__ anchors, float* __restrict__ ws) {
    const int li  = blockIdx.x;
    const int tid = threadIdx.x;
    const int W = (li == 0) ? 80 : (li == 1) ? 40 : 20;
    const int H = W;
    const float* preds = (li == 0) ? p0 : (li == 1) ? p1 : p2;
    float* tobj = ws + ((li == 0) ? TOBJ0 : (li == 1) ? TOBJ1 : TOBJ2);

    const int NA = 3, NT = 300, B = 16, C = 85, NC = 80;
    float lbox = 0.f, lcls = 0.f, nv = 0.f;

    if (tid < NA * NT) {
        const int a = tid / NT;
        const int j = tid % NT;
        const float* tg = targets + j * 6;
        const float gx = tg[2] * (float)W;
        const float gy = tg[3] * (float)H;
        const float gw = tg[4] * (float)W;
        const float gh = tg[5] * (float)H;
        const float ax = anchors[(li * 3 + a) * 2 + 0];
        const float ay = anchors[(li * 3 + a) * 2 + 1];
        const float rx = gw / ax, ry = gh / ay;
        const float m  = fmaxf(fmaxf(rx, 1.f / rx), fmaxf(ry, 1.f / ry));
        if (m < 4.f) {
            const int b = (int)tg[0];
            const int c = (int)tg[1];
            const int gi = min(max((int)gx, 0), W - 1);
            const int gj = min(max((int)gy, 0), H - 1);
            // NB: reference does anchors[a, a] (diagonal), not anchors[li, a].
            const float anx = anchors[a * 8 + 0];
            const float any_ = anchors[a * 8 + 1];
            const int match = B - 1 - b;
            const float* ps =
                preds + ((((size_t)match * NA + a) * H + gj) * (size_t)W + gi) * C;
            __builtin_prefetch(ps, 0, 0);                 // global_prefetch_b8
            __builtin_prefetch(ps + 64, 0, 0);            // rest of the 340B row

            const float px = 1.f / (1.f + expf(-ps[0]));
            const float py = 1.f / (1.f + expf(-ps[1]));
            const float pw = expf(ps[2]) * anx;
            const float ph = expf(ps[3]) * any_;
            const float tx = gx - truncf(gx);             // gij unclipped here
            const float ty = gy - truncf(gy);
            const float eps = 1e-9f;

            const float b1x1 = px - pw * 0.5f, b1x2 = px + pw * 0.5f;
            const float b1y1 = py - ph * 0.5f, b1y2 = py + ph * 0.5f;
            const float b2x1 = tx - gw * 0.5f, b2x2 = tx + gw * 0.5f;
            const float b2y1 = ty - gh * 0.5f, b2y2 = ty + gh * 0.5f;
            const float iw = fmaxf(fminf(b1x2, b2x2) - fmaxf(b1x1, b2x1), 0.f);
            const float ih = fmaxf(fminf(b1y2, b2y2) - fmaxf(b1y1, b2y1), 0.f);
            const float inter = iw * ih;
            const float w1 = b1x2 - b1x1, h1 = b1y2 - b1y1 + eps;
            const float w2 = b2x2 - b2x1, h2 = b2y2 - b2y1 + eps;
            const float uni = w1 * h1 + w2 * h2 - inter + eps;
            const float iou = inter / uni;
            const float cw = fmaxf(b1x2, b2x2) - fminf(b1x1, b2x1);
            const float ch = fmaxf(b1y2, b2y2) - fminf(b1y1, b2y1);
            const float c2 = cw * cw + ch * ch + eps;
            const float dxs = b2x1 + b2x2 - b1x1 - b1x2;
            const float dys = b2y1 + b2y2 - b1y1 - b1y2;
            const float rho2 = (dxs * dxs + dys * dys) * 0.25f;
            const float pi = 3.14159265358979323846f;
            const float dat = atanf(w2 / h2) - atanf(w1 / h1);
            const float v = (4.f / (pi * pi)) * dat * dat;
            const float alpha = v / (1.f + eps - iou + v);
            const float ciou = iou - (rho2 / c2 + v * alpha);

            lbox = 1.f - ciou;
            nv = 1.f;
            float s = 0.f;
            for (int k = 0; k < NC; ++k)
                s += bce_logits(ps[5 + k], (k == c) ? 1.f : 0.f);
            lcls = s;

            // tobj scatter: max() on float bits (values >= 0) for determinism
            const float tv = fmaxf(ciou, 0.f);
            const unsigned idx =
                (((unsigned)match * NA + a) * H + gj) * (unsigned)W + gi;
            atomicMax((unsigned int*)&tobj[idx], __float_as_uint(tv));
        }
    }

    __shared__ float red[3][1024];
    red[0][tid] = lbox; red[1][tid] = lcls; red[2][tid] = nv;
    __syncthreads();
    for (int s = 512; s > 0; s >>= 1) {
        if (tid < s) {
            red[0][tid] += red[0][tid + s];
            red[1][tid] += red[1][tid + s];
            red[2][tid] += red[2][tid + s];
        }
        __syncthreads();
    }
    if (tid == 0) {
        float* acc = ws + ACC + li * 3;
        acc[0] = red[0][0];
        acc[1] = red[1][0];
        acc[2] = red[2][0];
    }
}

// BCE over channel 4 of preds vs tobj; strided gather double-buffered through
// LDS with async loads (ASYNCcnt). Fixed per-block partials => deterministic.
__global__ __launch_bounds__(BLK)
void lobj_kernel(const float* __restrict__ preds, const float* __restrict__ tobj,
                 int N, float* __restrict__ partials) {
    __shared__ float sbuf[2 * BLK];
    const int tid = threadIdx.x;
    const int gid = blockIdx.x * BLK + tid;
    const int stride = (int)gridDim.x * BLK;
    const int iters = (N + stride - 1) / stride;

    int i0 = (gid < N) ? gid : (N - 1);
    async_ld_b32(preds + (size_t)i0 * 85 + 4, &sbuf[tid]);

    float acc = 0.f;
    int cur = 0;
    for (int k = 0; k < iters; ++k) {
        const int icur = gid + k * stride;
        if (k + 1 < iters) {                    // uniform across the grid
            int inext = gid + (k + 1) * stride;
            if (inext >= N) inext = N - 1;
            async_ld_b32(preds + (size_t)inext * 85 + 4,
                         &sbuf[(cur ^ 1) * BLK + tid]);
            wait_asynccnt<1>();                 // oldest transfer landed in LDS
        } else {
            wait_asynccnt<0>();
        }
        if (icur < N) {
            const float x = sbuf[cur * BLK + tid];
            const float t = tobj[icur];
            acc += bce_logits(x, t);
        }
        cur ^= 1;
    }

    __shared__ float red[BLK];
    red[tid] = acc;
    __syncthreads();
    for (int s = BLK / 2; s > 0; s >>= 1) {
        if (tid < s) red[tid] += red[tid + s];
        __syncthreads();
    }
    if (tid == 0) partials[blockIdx.x] = red[0];
}

__global__ void final_kernel(const float* __restrict__ ws, float* __restrict__ out) {
    const int Ns[3]    = {307200, 76800, 19200};
    const int pbase[3] = {P0OFF, P1OFF, P2OFF};
    const int pcnt[3]  = {256, 128, 64};
    float lbox_t = 0.f, lcls_t = 0.f, lobj_t = 0.f;
    for (int li = 0; li < 3; ++li) {
        const float* a = ws + ACC + li * 3;
        const float nv = fmaxf(a[2], 1.f);
        lbox_t += a[0] / nv;
        lcls_t += a[1] / (nv * 80.f);
        float s = 0.f;
        const float* p = ws + pbase[li];
        for (int k = 0; k < pcnt[li]; ++k) s += p[k];
        lobj_t += s / (float)Ns[li];
    }
    const float lbox = 0.05f * lbox_t;
    const float lobj = 1.0f * lobj_t;
    const float lcls = 0.5f * lcls_t;
    const float loss = lbox + lobj + lcls;
    out[0] = loss; out[1] = lbox; out[2] = lobj; out[3] = lcls; out[4] = loss;
}

extern "C" void kernel_launch(void* const* d_in, const int* in_sizes, int n_in,
                              void* d_out, int out_size, void* d_ws, size_t ws_size,
                              hipStream_t stream) {
    const float* p0 = (const float*)d_in[0];
    const float* p1 = (const float*)d_in[1];
    const float* p2 = (const float*)d_in[2];
    const float* tg = (const float*)d_in[3];
    const float* an = (const float*)d_in[4];
    float* ws  = (float*)d_ws;
    float* out = (float*)d_out;

    zero_kernel<<<256, 256, 0, stream>>>(ws, 403200);
    match_kernel<<<3, 1024, 0, stream>>>(p0, p1, p2, tg, an, ws);
    lobj_kernel<<<256, BLK, 0, stream>>>(p0, ws + TOBJ0, 307200, ws + P0OFF);
    lobj_kernel<<<128, BLK, 0, stream>>>(p1, ws + TOBJ1,  76800, ws + P1OFF);
    lobj_kernel<<< 64, BLK, 0, stream>>>(p2, ws + TOBJ2,  19200, ws + P2OFF);
    final_kernel<<<1, 1, 0, stream>>>(ws, out);
}